// GAT_85143431676087
// MI455X (gfx1250) — compile-verified
//
#include <hip/hip_runtime.h>
#include <math.h>

typedef float v2f __attribute__((ext_vector_type(2)));
typedef float v8f __attribute__((ext_vector_type(8)));

#define F_IN 128
#define NEG_SLOPE 0.2f

// ---------------- helpers ----------------

__device__ inline void atomicMaxFloat(float* addr, float val) {
    // monotonic int/uint mapping of IEEE floats -> native atomics, no CAS loop
    if (val >= 0.0f) atomicMax((int*)addr, __float_as_int(val));
    else             atomicMin((unsigned int*)addr, __float_as_uint(val));
}

__device__ inline void edge_sd(const int* __restrict__ ei, int E, long e, int& s, int& d) {
    if (e < (long)E) { s = ei[e]; d = ei[(long)E + e]; }
    else             { s = d = (int)(e - E); }          // appended self-loops
}

// ---------------- init: zero accumulator, reset segment max/denom ----------------

__global__ void k_init(float* __restrict__ O, float* __restrict__ mx,
                       float* __restrict__ dn, int n, int dout) {
    long idx = (long)blockIdx.x * blockDim.x + threadIdx.x;
    long tot = (long)n * dout;
    if (idx < tot) O[idx] = 0.0f;
    if (idx < n) { mx[idx] = -__builtin_inff(); dn[idx] = 0.0f; }
}

// ---------------- GEMM: H = X @ W via V_WMMA_F32_16X16X4_F32 ----------------
// one wave per 16x16 output tile; K-loop in steps of 4 (pure fp32 matrix path)

__global__ void __launch_bounds__(32)
k_gemm(const float* __restrict__ X, const float* __restrict__ W,
       float* __restrict__ H, int n, int din, int dout) {
    int lane = threadIdx.x;              // 0..31, full wave active (EXEC all 1s)
    int mt = blockIdx.x * 16;
    int nt = blockIdx.y * 16;
    int half = lane >> 4;                // 0: K pair {0,1}; 1: K pair {2,3}
    int l15  = lane & 15;
    int arow = mt + l15;
    if (arow >= n) arow = n - 1;         // clamp (n % 16 == 0 for this problem)
    int bcol = nt + l15;

    v8f acc = {};
    for (int k = 0; k < din; k += 4) {
        int ka = k + half * 2;
        // A 16x4: lane holds X[row, ka] and X[row, ka+1]  (8B-aligned pair)
        v2f a = *(const v2f*)(X + (long)arow * din + ka);
        // B 4x16: lane holds W[ka, col] and W[ka+1, col]
        v2f b;
        b.x = W[(long)ka * dout + bcol];
        b.y = W[(long)(ka + 1) * dout + bcol];
        acc = __builtin_amdgcn_wmma_f32_16x16x4_f32(
                  false, a, false, b, (short)0, acc, false, false);
    }
    // D 16x16 f32: VGPR v -> row (v + 8*half), col = l15
    #pragma unroll
    for (int v = 0; v < 8; ++v) {
        int row = mt + v + 8 * half;
        if (row < n) H[(long)row * dout + bcol] = acc[v];
    }
}

// ---------------- per-node attention logits: as = H.a_src, ad = H.a_dst ----------------

__global__ void k_alpha(const float* __restrict__ H, const float* __restrict__ a_s,
                        const float* __restrict__ a_d, float* __restrict__ as,
                        float* __restrict__ ad, int n, int dout) {
    int node = (int)(((long)blockIdx.x * blockDim.x + threadIdx.x) >> 5);
    int lane = threadIdx.x & 31;
    if (node >= n) return;
    float s = 0.f, d = 0.f;
    for (int f = lane; f < dout; f += 32) {
        float h = H[(long)node * dout + f];
        s += h * a_s[f];
        d += h * a_d[f];
    }
    #pragma unroll
    for (int off = 16; off; off >>= 1) {
        s += __shfl_xor(s, off, 32);
        d += __shfl_xor(d, off, 32);
    }
    if (lane == 0) { as[node] = s; ad[node] = d; }
}

// ---------------- edge passes ----------------

__global__ void k_edge_max(const int* __restrict__ ei, const float* __restrict__ as,
                           const float* __restrict__ ad, float* __restrict__ alpha,
                           float* __restrict__ mx, int E, int n) {
    long e = (long)blockIdx.x * blockDim.x + threadIdx.x;
    if (e >= (long)E + n) return;
    int s, d; edge_sd(ei, E, e, s, d);
    float a = as[s] + ad[d];
    a = (a > 0.f) ? a : NEG_SLOPE * a;           // leaky relu
    alpha[e] = a;
    atomicMaxFloat(&mx[d], a);
}

__global__ void k_edge_exp(const int* __restrict__ ei, float* __restrict__ alpha,
                           const float* __restrict__ mx, float* __restrict__ dn,
                           int E, int n) {
    long e = (long)blockIdx.x * blockDim.x + threadIdx.x;
    if (e >= (long)E + n) return;
    int s, d; edge_sd(ei, E, e, s, d);
    float v = expf(alpha[e] - mx[d]);
    alpha[e] = v;                                 // overwrite in place with exp
    atomicAdd(&dn[d], v);
}

__global__ void k_edge_coef(const int* __restrict__ ei, float* __restrict__ alpha,
                            const float* __restrict__ dn, int E, int n) {
    long e = (long)blockIdx.x * blockDim.x + threadIdx.x;
    if (e >= (long)E + n) return;
    int s, d; edge_sd(ei, E, e, s, d);
    alpha[e] = alpha[e] / dn[d];                  // softmax coefficient
}

// flat over (E+n)*dout: O[dst,f] += coef[e] * H[src,f]   (L2-resident working set)
__global__ void k_scatter(const int* __restrict__ ei, const float* __restrict__ coef,
                          const float* __restrict__ H, float* __restrict__ O,
                          int E, int n, int sh) {
    long idx = (long)blockIdx.x * blockDim.x + threadIdx.x;
    int dout = 1 << sh;
    long tot = ((long)E + n) << sh;
    if (idx >= tot) return;
    long e = idx >> sh;
    int  f = (int)(idx & (dout - 1));
    int s, d; edge_sd(ei, E, e, s, d);
    atomicAdd(&O[((long)d << sh) + f], coef[e] * H[((long)s << sh) + f]);
}

// ---------------- finalizers ----------------

__global__ void k_silu(const float* __restrict__ O, const float* __restrict__ b,
                       float* __restrict__ X, int n, int sh) {
    long idx = (long)blockIdx.x * blockDim.x + threadIdx.x;
    long tot = (long)n << sh;
    if (idx >= tot) return;
    int f = (int)(idx & ((1 << sh) - 1));
    float z = O[idx] + b[f];
    X[idx] = z / (1.0f + expf(-z));               // SiLU
}

// wave per row, C = 64 classes -> 2 values per lane
__global__ void k_logsoftmax(const float* __restrict__ O, const float* __restrict__ b,
                             float* __restrict__ out, int n) {
    int row  = (int)(((long)blockIdx.x * blockDim.x + threadIdx.x) >> 5);
    int lane = threadIdx.x & 31;
    if (row >= n) return;
    float z0 = O[(long)row * 64 + lane]      + b[lane];
    float z1 = O[(long)row * 64 + lane + 32] + b[lane + 32];
    float m = fmaxf(z0, z1);
    #pragma unroll
    for (int off = 16; off; off >>= 1) m = fmaxf(m, __shfl_xor(m, off, 32));
    float s = expf(z0 - m) + expf(z1 - m);
    #pragma unroll
    for (int off = 16; off; off >>= 1) s += __shfl_xor(s, off, 32);
    float lse = m + logf(s);
    out[(long)row * 64 + lane]      = z0 - lse;
    out[(long)row * 64 + lane + 32] = z1 - lse;
}

// ---------------- orchestration ----------------

static void run_layer(const float* X, const float* W, const float* avs, const float* avd,
                      float* H, float* O, float* ecoef, float* asv, float* adv,
                      float* mxv, float* dnv, const int* ei,
                      int n, int E, int din, int dout, hipStream_t stream) {
    int sh = (dout == 128) ? 7 : 6;
    long nTot  = (long)n * dout;
    long eTot  = (long)E + n;
    long sTot  = eTot << sh;

    k_init<<<(unsigned)((nTot + 255) / 256), 256, 0, stream>>>(O, mxv, dnv, n, dout);

    dim3 gg((n + 15) / 16, dout / 16);
    k_gemm<<<gg, 32, 0, stream>>>(X, W, H, n, din, dout);

    k_alpha<<<(unsigned)(((long)n * 32 + 127) / 128), 128, 0, stream>>>(H, avs, avd, asv, adv, n, dout);

    k_edge_max<<<(unsigned)((eTot + 255) / 256), 256, 0, stream>>>(ei, asv, adv, ecoef, mxv, E, n);
    k_edge_exp<<<(unsigned)((eTot + 255) / 256), 256, 0, stream>>>(ei, ecoef, mxv, dnv, E, n);
    k_edge_coef<<<(unsigned)((eTot + 255) / 256), 256, 0, stream>>>(ei, ecoef, dnv, E, n);

    k_scatter<<<(unsigned)((sTot + 255) / 256), 256, 0, stream>>>(ei, ecoef, H, O, E, n, sh);
}

extern "C" void kernel_launch(void* const* d_in, const int* in_sizes, int n_in,
                              void* d_out, int out_size, void* d_ws, size_t ws_size,
                              hipStream_t stream) {
    const float* x   = (const float*)d_in[0];
    const int*   ei  = (const int*)  d_in[1];
    const float* W0  = (const float*)d_in[2];
    const float* aS0 = (const float*)d_in[3];
    const float* aD0 = (const float*)d_in[4];
    const float* b0  = (const float*)d_in[5];
    const float* W1  = (const float*)d_in[6];
    const float* aS1 = (const float*)d_in[7];
    const float* aD1 = (const float*)d_in[8];
    const float* b1  = (const float*)d_in[9];
    const float* W2  = (const float*)d_in[10];
    const float* aS2 = (const float*)d_in[11];
    const float* aD2 = (const float*)d_in[12];
    const float* b2  = (const float*)d_in[13];
    float* out = (float*)d_out;

    const int n = in_sizes[0] / F_IN;     // 50000
    const int E = in_sizes[1] / 2;        // 600000

    float* ws = (float*)d_ws;
    size_t off = 0;
    float* B0    = ws + off; off += (size_t)n * 128;
    float* B1    = ws + off; off += (size_t)n * 128;
    float* B2    = ws + off; off += (size_t)n * 128;
    float* ecoef = ws + off; off += (size_t)(E + n);
    float* asv   = ws + off; off += (size_t)n;
    float* adv   = ws + off; off += (size_t)n;
    float* mxv   = ws + off; off += (size_t)n;
    float* dnv   = ws + off; off += (size_t)n;

    // Layer 0: X = x, H = B0, O = B1; SiLU(O+b0) -> B0
    run_layer(x,  W0, aS0, aD0, B0, B1, ecoef, asv, adv, mxv, dnv, ei, n, E, 128, 128, stream);
    k_silu<<<(unsigned)(((long)n * 128 + 255) / 256), 256, 0, stream>>>(B1, b0, B0, n, 7);

    // Layer 1: X = B0, H = B1, O = B2; SiLU(O+b1) -> B1
    run_layer(B0, W1, aS1, aD1, B1, B2, ecoef, asv, adv, mxv, dnv, ei, n, E, 128, 128, stream);
    k_silu<<<(unsigned)(((long)n * 128 + 255) / 256), 256, 0, stream>>>(B2, b1, B1, n, 7);

    // Layer 2: X = B1, H = B2, O = B0 (n*64); log_softmax(O+b2) -> d_out
    run_layer(B1, W2, aS2, aD2, B2, B0, ecoef, asv, adv, mxv, dnv, ei, n, E, 128, 64, stream);
    k_logsoftmax<<<(unsigned)(((long)n * 32 + 127) / 128), 128, 0, stream>>>(B0, b2, out, n);
}